// RWKV_RNN_22806276342270
// MI455X (gfx1250) — compile-verified
//
#include <hip/hip_runtime.h>

#define L_   12
#define D_   1024
#define H_   16
#define S_   64
#define FF_  3584
#define TPB  256
#define NWAVE 8

typedef float v2f __attribute__((ext_vector_type(2)));
typedef float v8f __attribute__((ext_vector_type(8)));

// ---------------------------------------------------------------------------
// Core: y[16] (one 16-row tile) = W_tile @ x  via V_WMMA_F32_16X16X4_F32.
// A (16x4 f32): lane = row M, VGPR0/1 = K pair, lane-half selects K{0,1}/{2,3}.
// B (4x16 f32): every column set equal to x chunk -> all D columns equal y.
// 8 waves split K; deterministic LDS reduction. Valid return only for tid<16.
// Weights are streamed non-temporally (one-pass 640 MB); compiler emits an
// 8-deep global_load_b64 NT clause + ds_load_2addr_b64 + 8 chained WMMAs.
// ---------------------------------------------------------------------------
__device__ __forceinline__ float mv16_wmma(const float* __restrict__ Wt, int K,
                                           const float* xs, float* red)
{
  const int tid  = threadIdx.x;
  const int wave = tid >> 5;
  const int lane = tid & 31;
  const int half = lane >> 4;          // 0: K{0,1}, 1: K{2,3}
  const int mrow = lane & 15;          // output row within tile
  const int kslice = K / NWAVE;        // multiple of 4
  const int kb = wave * kslice;
  const int ke = kb + kslice;
  const float* wrow = Wt + (size_t)mrow * K + 2 * half;
  v8f acc = {};
  for (int k0 = kb; k0 < ke; k0 += 4) {
    v2f a = __builtin_nontemporal_load((const v2f*)(wrow + k0));
    v2f b;
    b.x = xs[k0 + 2 * half];
    b.y = xs[k0 + 2 * half + 1];
    acc = __builtin_amdgcn_wmma_f32_16x16x4_f32(false, a, false, b,
                                                (short)0, acc, false, false);
  }
  // D layout: lane 0 holds rows 0..7 (VGPR i = row i), lane 16 rows 8..15.
  if (lane == 0) {
#pragma unroll
    for (int i = 0; i < 8; ++i) red[wave * 16 + i] = acc[i];
  } else if (lane == 16) {
#pragma unroll
    for (int i = 0; i < 8; ++i) red[wave * 16 + 8 + i] = acc[i];
  }
  __syncthreads();
  float s = 0.f;
  if (tid < 16) {
#pragma unroll
    for (int w = 0; w < NWAVE; ++w) s += red[w * 16 + tid];
  }
  return s;
}

__device__ __forceinline__ void stage(const float* __restrict__ g, float* s, int n) {
  for (int i = threadIdx.x; i < n; i += TPB) s[i] = g[i];
  __syncthreads();
}

// ---------------------------------------------------------------------------
// Embedding row + LN0
// ---------------------------------------------------------------------------
__global__ void k_embed_ln(const int* __restrict__ tok, const float* __restrict__ emb,
                           const float* __restrict__ w, const float* __restrict__ b,
                           float* __restrict__ x)
{
  __shared__ float s1[TPB], s2[TPB];
  const float* row = emb + (size_t)tok[0] * D_;
  int tid = threadIdx.x;
  float a0 = 0.f, a1 = 0.f;
  for (int i = tid; i < D_; i += TPB) { float t = row[i]; a0 += t; a1 += t * t; }
  s1[tid] = a0; s2[tid] = a1; __syncthreads();
  for (int off = TPB / 2; off > 0; off >>= 1) {
    if (tid < off) { s1[tid] += s1[tid + off]; s2[tid] += s2[tid + off]; }
    __syncthreads();
  }
  float m  = s1[0] * (1.f / D_);
  float v  = s2[0] * (1.f / D_) - m * m;
  float rs = rsqrtf(v + 1e-5f);
  for (int i = tid; i < D_; i += TPB) x[i] = (row[i] - m) * rs * w[i] + b[i];
}

// ---------------------------------------------------------------------------
// LN + 4-way time-mix (attention input vectors)
// ---------------------------------------------------------------------------
__global__ void k_ln_mix4(const float* __restrict__ x, const float* __restrict__ w,
                          const float* __restrict__ b,
                          const float* __restrict__ tk, const float* __restrict__ tv,
                          const float* __restrict__ tr, const float* __restrict__ tg,
                          const float* __restrict__ sa,
                          float* __restrict__ xk, float* __restrict__ xv,
                          float* __restrict__ xr, float* __restrict__ xg)
{
  __shared__ float s1[TPB], s2[TPB];
  int tid = threadIdx.x;
  float a0 = 0.f, a1 = 0.f;
  for (int i = tid; i < D_; i += TPB) { float t = x[i]; a0 += t; a1 += t * t; }
  s1[tid] = a0; s2[tid] = a1; __syncthreads();
  for (int off = TPB / 2; off > 0; off >>= 1) {
    if (tid < off) { s1[tid] += s1[tid + off]; s2[tid] += s2[tid + off]; }
    __syncthreads();
  }
  float m = s1[0] * (1.f / D_), v = s2[0] * (1.f / D_) - m * m;
  float rs = rsqrtf(v + 1e-5f);
  for (int i = tid; i < D_; i += TPB) {
    float xln = (x[i] - m) * rs * w[i] + b[i];
    float sv  = sa[i];
    xk[i] = xln * tk[i] + sv * (1.f - tk[i]);
    xv[i] = xln * tv[i] + sv * (1.f - tv[i]);
    xr[i] = xln * tr[i] + sv * (1.f - tr[i]);
    xg[i] = xln * tg[i] + sv * (1.f - tg[i]);
  }
}

// ---------------------------------------------------------------------------
// LN + 2-way time-mix (ffn input vectors)
// ---------------------------------------------------------------------------
__global__ void k_ln_mix2(const float* __restrict__ x, const float* __restrict__ w,
                          const float* __restrict__ b,
                          const float* __restrict__ tk, const float* __restrict__ tr,
                          const float* __restrict__ sf,
                          float* __restrict__ xk, float* __restrict__ xr)
{
  __shared__ float s1[TPB], s2[TPB];
  int tid = threadIdx.x;
  float a0 = 0.f, a1 = 0.f;
  for (int i = tid; i < D_; i += TPB) { float t = x[i]; a0 += t; a1 += t * t; }
  s1[tid] = a0; s2[tid] = a1; __syncthreads();
  for (int off = TPB / 2; off > 0; off >>= 1) {
    if (tid < off) { s1[tid] += s1[tid + off]; s2[tid] += s2[tid + off]; }
    __syncthreads();
  }
  float m = s1[0] * (1.f / D_), v = s2[0] * (1.f / D_) - m * m;
  float rs = rsqrtf(v + 1e-5f);
  for (int i = tid; i < D_; i += TPB) {
    float xln = (x[i] - m) * rs * w[i] + b[i];
    float sv  = sf[i];
    xk[i] = xln * tk[i] + sv * (1.f - tk[i]);
    xr[i] = xln * tr[i] + sv * (1.f - tr[i]);
  }
}

// ---------------------------------------------------------------------------
// r/k/v/g matvecs (4 x [1024x1024]); g gets SiLU. grid = 4*64 blocks.
// ---------------------------------------------------------------------------
__global__ void k_mv_rkvg(const float* __restrict__ kw, const float* __restrict__ vw,
                          const float* __restrict__ rw, const float* __restrict__ gw,
                          const float* __restrict__ xk, const float* __restrict__ xv,
                          const float* __restrict__ xr, const float* __restrict__ xg,
                          float* __restrict__ ko, float* __restrict__ vo,
                          float* __restrict__ ro, float* __restrict__ go)
{
  __shared__ float xs[D_];
  __shared__ float red[NWAVE * 16];
  int mat  = blockIdx.x >> 6;
  int tile = blockIdx.x & 63;
  const float* W; const float* xin; float* out;
  if (mat == 0)      { W = kw; xin = xk; out = ko; }
  else if (mat == 1) { W = vw; xin = xv; out = vo; }
  else if (mat == 2) { W = rw; xin = xr; out = ro; }
  else               { W = gw; xin = xg; out = go; }
  stage(xin, xs, D_);
  float s = mv16_wmma(W + (size_t)tile * 16 * D_, D_, xs, red);
  if (threadIdx.x < 16) {
    int m = tile * 16 + threadIdx.x;
    if (mat == 3) s = s / (1.f + __expf(-s));   // SiLU
    out[m] = s;
  }
}

// ---------------------------------------------------------------------------
// wkv readout (+groupnorm, +gate) recomputed per block (state is L2-resident),
// then one 16-row tile of ow @ gg, with residual. grid = 64 blocks.
// wkv[h,d] = v[h,d]*sum_s(r*tf*k) + r[h,:] . state[h,:,d]   (new state unused)
// ---------------------------------------------------------------------------
__global__ void k_wkv_ow(const float* __restrict__ rb, const float* __restrict__ kb,
                         const float* __restrict__ vb, const float* __restrict__ gb,
                         const float* __restrict__ tf, const float* __restrict__ st,
                         const float* __restrict__ lxw, const float* __restrict__ lxb,
                         const float* __restrict__ ow, float* __restrict__ x)
{
  __shared__ float rsh[D_];
  __shared__ float wk[D_];
  __shared__ float gg[D_];
  __shared__ float c0[H_], hm[H_], hr[H_];
  __shared__ float red[NWAVE * 16];
  int tid = threadIdx.x;
  for (int i = tid; i < D_; i += TPB) rsh[i] = rb[i];
  __syncthreads();
  if (tid < H_) {
    float c = 0.f;
    for (int s2 = 0; s2 < S_; ++s2) {
      int idx = tid * S_ + s2;
      c += rsh[idx] * tf[idx] * kb[idx];
    }
    c0[tid] = c;
  }
  __syncthreads();
  for (int d = tid; d < D_; d += TPB) {
    int h = d >> 6;
    float acc = c0[h] * vb[d];
    const float* stc = st + (size_t)h * S_ * S_ + (d & 63);
    const float* rh  = rsh + h * S_;
    for (int s2 = 0; s2 < S_; ++s2) acc += rh[s2] * stc[s2 * S_];
    wk[d] = acc;
  }
  __syncthreads();
  if (tid < H_) {
    float m = 0.f, v = 0.f;
    for (int s2 = 0; s2 < S_; ++s2) m += wk[tid * S_ + s2];
    m *= (1.f / S_);
    for (int s2 = 0; s2 < S_; ++s2) { float t = wk[tid * S_ + s2] - m; v += t * t; }
    v *= (1.f / S_);
    hm[tid] = m; hr[tid] = rsqrtf(v + 1e-5f);
  }
  __syncthreads();
  for (int d = tid; d < D_; d += TPB) {
    int h = d >> 6;
    gg[d] = ((wk[d] - hm[h]) * hr[h] * lxw[d] + lxb[d]) * gb[d];
  }
  __syncthreads();
  int tile = blockIdx.x;
  float s = mv16_wmma(ow + (size_t)tile * 16 * D_, D_, gg, red);
  if (tid < 16) { int m = tile * 16 + tid; x[m] += s; }
}

// ---------------------------------------------------------------------------
// ffn: kk = relu(fkw@xk2)^2 (224 tiles) and rr = sigmoid(frw@xr2) (64 tiles).
// grid = 288 blocks.
// ---------------------------------------------------------------------------
__global__ void k_ffn_kr(const float* __restrict__ fkw, const float* __restrict__ frw,
                         const float* __restrict__ xk2, const float* __restrict__ xr2,
                         float* __restrict__ kk, float* __restrict__ rr)
{
  __shared__ float xs[D_];
  __shared__ float red[NWAVE * 16];
  int b = blockIdx.x;
  bool isK = (b < (FF_ / 16));
  int tile = isK ? b : (b - FF_ / 16);
  const float* W   = isK ? fkw : frw;
  const float* xin = isK ? xk2 : xr2;
  stage(xin, xs, D_);
  float s = mv16_wmma(W + (size_t)tile * 16 * D_, D_, xs, red);
  if (threadIdx.x < 16) {
    int m = tile * 16 + threadIdx.x;
    if (isK) { float t = fmaxf(s, 0.f); kk[m] = t * t; }
    else     { rr[m] = 1.f / (1.f + __expf(-s)); }
  }
}

// ---------------------------------------------------------------------------
// x += rr * (fvw @ kk); K = 3584. grid = 64 blocks.
// ---------------------------------------------------------------------------
__global__ void k_ffn_v(const float* __restrict__ fvw, const float* __restrict__ kk,
                        const float* __restrict__ rr, float* __restrict__ x)
{
  __shared__ float xs[FF_];
  __shared__ float red[NWAVE * 16];
  stage(kk, xs, FF_);
  int tile = blockIdx.x;
  float s = mv16_wmma(fvw + (size_t)tile * 16 * FF_, FF_, xs, red);
  if (threadIdx.x < 16) { int m = tile * 16 + threadIdx.x; x[m] += rr[m] * s; }
}

__global__ void k_copy(const float* __restrict__ src, float* __restrict__ dst) {
  int i = blockIdx.x * TPB + threadIdx.x;
  if (i < D_) dst[i] = src[i];
}

// ---------------------------------------------------------------------------
extern "C" void kernel_launch(void* const* d_in, const int* in_sizes, int n_in,
                              void* d_out, int out_size, void* d_ws, size_t ws_size,
                              hipStream_t stream) {
  const int*   tok  = (const int*)d_in[0];
  const float* emb  = (const float*)d_in[1];
  const float* ln0w = (const float*)d_in[2],  *ln0b = (const float*)d_in[3];
  const float* ln1w = (const float*)d_in[4],  *ln1b = (const float*)d_in[5];
  const float* ln2w = (const float*)d_in[6],  *ln2b = (const float*)d_in[7];
  const float* tmk  = (const float*)d_in[8],  *tmv  = (const float*)d_in[9];
  const float* tmr  = (const float*)d_in[10], *tmg  = (const float*)d_in[11];
  const float* tf   = (const float*)d_in[12];
  /* d_in[13] att_time_decay: unused — new_wkv state is discarded by reference */
  const float* kw   = (const float*)d_in[14], *vw   = (const float*)d_in[15];
  const float* rw   = (const float*)d_in[16], *gw   = (const float*)d_in[17];
  const float* ow   = (const float*)d_in[18];
  const float* lxw  = (const float*)d_in[19], *lxb  = (const float*)d_in[20];
  const float* fmk  = (const float*)d_in[21], *fmr  = (const float*)d_in[22];
  const float* fkw  = (const float*)d_in[23], *fvw  = (const float*)d_in[24];
  const float* frw  = (const float*)d_in[25];
  const float* sat  = (const float*)d_in[26];
  const float* stw  = (const float*)d_in[27];
  const float* sff  = (const float*)d_in[28];

  float* ws  = (float*)d_ws;
  float* X   = ws;
  float* XK  = ws + 1024;  float* XV  = ws + 2048;
  float* XR  = ws + 3072;  float* XG  = ws + 4096;
  float* Rb  = ws + 5120;  float* Kb  = ws + 6144;
  float* Vb  = ws + 7168;  float* Gb  = ws + 8192;
  float* XK2 = ws + 9216;  float* XR2 = ws + 10240;
  float* RR  = ws + 11264; float* KK  = ws + 12288;   // 3584 floats

  k_embed_ln<<<1, TPB, 0, stream>>>(tok, emb, ln0w, ln0b, X);
  for (int l = 0; l < L_; ++l) {
    size_t oD = (size_t)l * D_, oDD = (size_t)l * D_ * D_, oHS = (size_t)l * H_ * S_;
    k_ln_mix4<<<1, TPB, 0, stream>>>(X, ln1w + oD, ln1b + oD,
                                     tmk + oD, tmv + oD, tmr + oD, tmg + oD,
                                     sat + oD, XK, XV, XR, XG);
    k_mv_rkvg<<<256, TPB, 0, stream>>>(kw + oDD, vw + oDD, rw + oDD, gw + oDD,
                                       XK, XV, XR, XG, Kb, Vb, Rb, Gb);
    k_wkv_ow<<<64, TPB, 0, stream>>>(Rb, Kb, Vb, Gb, tf + oHS,
                                     stw + (size_t)l * H_ * S_ * S_,
                                     lxw + oD, lxb + oD, ow + oDD, X);
    k_ln_mix2<<<1, TPB, 0, stream>>>(X, ln2w + oD, ln2b + oD,
                                     fmk + oD, fmr + oD, sff + oD, XK2, XR2);
    k_ffn_kr<<<288, TPB, 0, stream>>>(fkw + (size_t)l * FF_ * D_, frw + oDD,
                                      XK2, XR2, KK, RR);
    k_ffn_v<<<64, TPB, 0, stream>>>(fvw + (size_t)l * D_ * FF_, KK, RR, X);
  }
  k_copy<<<4, TPB, 0, stream>>>(X, (float*)d_out);
  (void)in_sizes; (void)n_in; (void)out_size; (void)ws_size;
}